// Vocab_Position_Tokentype_ParallelEmbedding_35862976921834
// MI455X (gfx1250) — compile-verified
//
#include <hip/hip_runtime.h>

// Problem constants (match reference).
#define BATCH   8
#define SEQ     2048
#define HIDDEN  1024

typedef float v4f __attribute__((ext_vector_type(4)));

// One block per sequence position s. 256 threads (8 wave32) cover H=1024 floats
// as float4 chunks. pos row + both tokentype rows are loaded once and reused
// across all BATCH=8 gathers; vocab rows are prefetched (global_prefetch_b8)
// before the unrolled gather/add/store loop; output uses non-temporal B128
// stores (written once, never re-read on device).
__global__ __launch_bounds__(256, 4)
void vocab_pos_tt_embed_kernel(const int* __restrict__ idx,
                               const int* __restrict__ types,
                               const float* __restrict__ vocab,   // [VOCAB, H]
                               const float* __restrict__ pos,     // [SEQ, H]
                               const float* __restrict__ tt,      // [2, H]
                               float* __restrict__ out)           // [B, SEQ, H]
{
    const int s = blockIdx.x;          // 0..SEQ-1
    const int c = threadIdx.x * 4;     // column chunk: 4 floats per thread

    // Block-uniform token / type ids for all 8 batches (s_load path).
    int tok[BATCH];
    int typ[BATCH];
#pragma unroll
    for (int b = 0; b < BATCH; ++b) {
        tok[b] = idx[b * SEQ + s];
        typ[b] = types[b * SEQ + s];
        // Kick the random vocab row toward L2/L1 early: gfx1250 global_prefetch_b8.
        __builtin_prefetch(vocab + (size_t)tok[b] * HIDDEN + c, /*rw=*/0, /*loc=*/3);
    }

    // Shared addends, loaded once per block, reused 8x.
    const v4f p  = *(const v4f*)(pos + (size_t)s * HIDDEN + c);
    const v4f t0 = *(const v4f*)(tt + 0 * HIDDEN + c);
    const v4f t1 = *(const v4f*)(tt + 1 * HIDDEN + c);

#pragma unroll
    for (int b = 0; b < BATCH; ++b) {
        const v4f v = *(const v4f*)(vocab + (size_t)tok[b] * HIDDEN + c);
        const v4f t = (typ[b] != 0) ? t1 : t0;   // v_cndmask per lane
        const v4f r = v + p + t;
        // Streamed output: non-temporal store, don't pollute L2 with 64 MiB.
        __builtin_nontemporal_store(r, (v4f*)(out + ((size_t)(b * SEQ + s)) * HIDDEN + c));
    }
}

extern "C" void kernel_launch(void* const* d_in, const int* in_sizes, int n_in,
                              void* d_out, int out_size, void* d_ws, size_t ws_size,
                              hipStream_t stream)
{
    (void)in_sizes; (void)n_in; (void)out_size; (void)d_ws; (void)ws_size;

    const int*   idx   = (const int*)  d_in[0];  // [B, SEQ] int32
    const int*   types = (const int*)  d_in[1];  // [B, SEQ] int32
    const float* vocab = (const float*)d_in[2];  // [VOCAB, H] f32
    const float* pos   = (const float*)d_in[3];  // [SEQ, H] f32
    const float* tt    = (const float*)d_in[4];  // [2, H] f32
    float*       out   = (float*)      d_out;    // [B, SEQ, H] f32

    dim3 grid(SEQ);                 // 2048 blocks, one per sequence position
    dim3 block(HIDDEN / 4);         // 256 threads = 8 wave32
    vocab_pos_tt_embed_kernel<<<grid, block, 0, stream>>>(idx, types, vocab, pos, tt, out);
}